// RoiTrainingModel_80934363725970
// MI455X (gfx1250) — compile-verified
//
#include <hip/hip_runtime.h>
#include <hip/hip_bf16.h>
#include <stdint.h>

// ---------------------------------------------------------------------------
// Faster R-CNN ROI sampling + loss for MI455X (gfx1250, wave32).
//
// Phase A (782 blocks x 256): IoU(200000 ROIs x 128 GT) with the GT table
//          staged into LDS by the Tensor Data Mover (tensor_load_to_lds,
//          s_wait_tensorcnt).  Per-ROI: max-IoU, argmax GT, pos/neg label,
//          deterministic hash key for sampling.
// Phase B (1 block x 1024): deterministic top-32 (pos) / top-128 (neg)
//          selection via iterative arg-max with a strictly decreasing key
//          threshold (keys unique -> no marking, no atomics).
// Phase C (1 block x 256 = 8 waves): per-sample log-softmax CE across 81
//          classes spread over 32 lanes (wave32 butterfly reductions) +
//          smooth-L1 box regression for positives; single deterministic
//          reduction writes d_out[0]=cls_loss, d_out[1]=reg_loss.
//
// Workspace: poskey u64[200000] | negkey u64[200000] | gtidx u32[200000]
//            | sel i32[160] | meta i32[1]   (~4.0 MB)
// ---------------------------------------------------------------------------

#define N_ROIS        200000
#define N_GT          128
#define N_CLS         81
#define MAX_POS       32
#define TOTAL_SAMPLES 128
#define N_SEL         (MAX_POS + TOTAL_SAMPLES)   // 160

#ifndef __has_builtin
#define __has_builtin(x) 0
#endif
#if __has_builtin(__builtin_amdgcn_tensor_load_to_lds) && \
    __has_builtin(__builtin_amdgcn_s_wait_tensorcnt)
#define USE_TDM 1
#else
#define USE_TDM 0
#endif

typedef unsigned int       u32x4 __attribute__((ext_vector_type(4)));
typedef int                i32x8 __attribute__((ext_vector_type(8)));
typedef int                i32x4 __attribute__((ext_vector_type(4)));
typedef unsigned long long u64;

__device__ __forceinline__ unsigned mix32(unsigned x) {
  x ^= x >> 16; x *= 0x85ebca6bu;
  x ^= x >> 13; x *= 0xc2b2ae35u;
  x ^= x >> 16;
  return x;
}

__device__ __forceinline__ u64 umax64(u64 a, u64 b) { return a > b ? a : b; }

// ---------------------------------------------------------------------------
// Phase A: IoU + labeling + sampling keys
// ---------------------------------------------------------------------------
__global__ __launch_bounds__(256) void iou_label_kernel(
    const float* __restrict__ rois, const float* __restrict__ gt,
    u64* __restrict__ poskey, u64* __restrict__ negkey,
    unsigned* __restrict__ gtidx) {
  // ONLY static LDS object in this kernel -> allocated at LDS offset 0,
  // which is what the TDM descriptor's lds_addr field assumes.
  __shared__ float s_gt[N_GT * 4];

#if USE_TDM
  if (threadIdx.x == 0) {
    // Build Tensor DMA Descriptor (D#) per CDNA5 ISA 8.3/8.4:
    // 1-D tensor of 512 f32 (the whole GT table), tile = whole tensor.
    unsigned long long ga = (unsigned long long)(const void*)gt;
    u32x4 g0 = {
        1u,                                   // count=1 (valid), user mode
        0u,                                   // lds_addr = 0 (s_gt)
        (unsigned)(ga & 0xFFFFFFFFull),       // global_addr[31:0]
        ((unsigned)((ga >> 32) & 0x1FFFFFFull)) | (2u << 30)  // [56:32] | type=2
    };
    i32x8 g1 = {
        (int)(2u << 16),            // workgroup_mask=0 (not in cluster), data_size=4B
        (int)(512u << 16),          // atomic_barrier=0 ; tensor_dim0[15:0]=512
        (int)(1u << 16),            // tensor_dim0[31:16]=0 ; tensor_dim1=1
        (int)(512u << 16),          // tensor_dim1 hi=0 ; tile_dim0=512
        0,                          // tile_dim1=0, tile_dim2=0 (unused)
        512,                        // tensor_dim0_stride[31:0]
        0, 0                        // stride hi / tensor_dim1_stride = 0
    };
    i32x4 z4 = {0, 0, 0, 0};                    // groups 2/3 unused (<=2D tensor)
    i32x8 z8 = {0, 0, 0, 0, 0, 0, 0, 0};        // trailing group (clang-23 6-arg form)
    __builtin_amdgcn_tensor_load_to_lds(g0, g1, z4, z4, z8, 0);
    __builtin_amdgcn_s_wait_tensorcnt(0);
  }
#else
  for (int t = threadIdx.x; t < N_GT * 4; t += blockDim.x) s_gt[t] = gt[t];
#endif
  __syncthreads();

  const int i = blockIdx.x * 256 + threadIdx.x;
  if (i >= N_ROIS) return;

  const float4 rb = reinterpret_cast<const float4*>(rois)[i];  // b128 load
  const float area_r = (rb.z - rb.x) * (rb.w - rb.y);

  // Track best IoU by cross-multiplication (unions > 0) -> one divide total.
  float bi = 0.0f, bd = 1.0f;
  unsigned bidx = 0;
  const float4* gt4 = reinterpret_cast<const float4*>(s_gt);
#pragma unroll 4
  for (int m = 0; m < N_GT; ++m) {
    const float4 g = gt4[m];                 // LDS b128 broadcast
    const float area_g = (g.z - g.x) * (g.w - g.y);
    float w = fminf(rb.z, g.z) - fmaxf(rb.x, g.x);
    float h = fminf(rb.w, g.w) - fmaxf(rb.y, g.y);
    w = fmaxf(w, 0.0f);
    h = fmaxf(h, 0.0f);
    const float inter = w * h;
    const float uni = fmaxf(area_r + area_g - inter, 1e-8f);
    if (inter * bd > bi * uni) { bi = inter; bd = uni; bidx = (unsigned)m; }
  }
  const float max_iou = bi / bd;

  int label = -1;
  if (max_iou > 0.5f) label = 1;
  if (max_iou < 0.1f) label = 0;

  // Deterministic per-ROI uniform key (stand-in for jax.random; threefry is
  // not reproducible bit-exactly here).  63-bit key, unique via low word.
  const unsigned h32 = mix32((unsigned)i * 2654435761u);
  const u64 key = ((u64)(h32 & 0x7FFFFFFFu) << 32) | (u64)(0xFFFFFFFFu - (unsigned)i);
  poskey[i] = (label == 1) ? key : 0ull;
  negkey[i] = (label == 0) ? key : 0ull;
  gtidx[i] = bidx;
}

// ---------------------------------------------------------------------------
// Phase B: deterministic top-k selection (iterative arg-max w/ threshold)
// ---------------------------------------------------------------------------
__global__ __launch_bounds__(1024) void select_kernel(
    const u64* __restrict__ poskey, const u64* __restrict__ negkey,
    int* __restrict__ sel, int* __restrict__ meta) {
  __shared__ u64 smax[1024];
  const int tid = threadIdx.x;

  const u64* keys = poskey;
  int base = 0, count = MAX_POS;
  for (int phase = 0; phase < 2; ++phase) {
    u64 thr = ~0ull;  // exclusive upper bound; keys < 2^63 so all qualify
    for (int j = 0; j < count; ++j) {
      u64 lm = 0;
      for (int i = tid; i < N_ROIS; i += 1024) {
        __builtin_prefetch(keys + i + 4096, 0, 1);  // global_prefetch_b8
        const u64 k = keys[i];
        lm = (k < thr && k > lm) ? k : lm;
      }
      smax[tid] = lm;
      __syncthreads();
      for (int s = 512; s > 0; s >>= 1) {
        if (tid < s) smax[tid] = umax64(smax[tid], smax[tid + s]);
        __syncthreads();
      }
      const u64 w = smax[0];
      __syncthreads();
      if (tid == 0)
        sel[base + j] =
            (w != 0ull) ? (int)(0xFFFFFFFFu - (unsigned)(w & 0xFFFFFFFFull)) : -1;
      thr = w;  // keys unique -> strictly-less threshold skips all selected
    }
    keys = negkey;
    base = MAX_POS;
    count = TOTAL_SAMPLES;
  }

  if (tid == 0) {
    int c = 0;
    for (int j = 0; j < MAX_POS; ++j) c += (sel[j] >= 0);
    meta[0] = c;  // cur_pos
  }
}

// ---------------------------------------------------------------------------
// Phase C: losses (one sample per wave32; lanes cover the 81 classes)
// ---------------------------------------------------------------------------
__global__ __launch_bounds__(256) void loss_kernel(
    const float* __restrict__ rois, const float* __restrict__ gt,
    const float* __restrict__ score, const float* __restrict__ pred,
    const int* __restrict__ gt_labels, const unsigned* __restrict__ gtidx,
    const int* __restrict__ sel, const int* __restrict__ meta,
    float* __restrict__ out) {
  __shared__ float partial[24];  // [0..7]=cls, [8..15]=reg, [16..23]=nvalid
  const int lane = threadIdx.x & 31;
  const int wave = threadIdx.x >> 5;  // 8 waves (wave32)
  const int cur_pos = meta[0];

  float cls_acc = 0.0f, reg_acc = 0.0f, nv = 0.0f;

  for (int s = wave; s < N_SEL; s += 8) {
    const int roi = sel[s];
    const bool slot_sel = (roi >= 0);
    bool valid;
    if (s < MAX_POS)
      valid = slot_sel;                                            // pos_valid
    else
      valid = slot_sel && ((s - MAX_POS) < TOTAL_SAMPLES - cur_pos);  // neg_valid
    if (!valid) continue;  // uniform per wave -> no shuffle divergence

    // ---- softmax cross-entropy over 81 classes, 32 lanes ----
    const float* row = score + (size_t)roi * N_CLS;
    const float x0 = row[lane];
    const float x1 = row[lane + 32];
    const float x2 = (lane < N_CLS - 64) ? row[lane + 64] : -1e30f;
    float m = fmaxf(fmaxf(x0, x1), x2);
    for (int o = 16; o > 0; o >>= 1) m = fmaxf(m, __shfl_xor(m, o, 32));
    float e = __expf(x0 - m) + __expf(x1 - m) +
              ((lane < N_CLS - 64) ? __expf(x2 - m) : 0.0f);
    for (int o = 16; o > 0; o >>= 1) e += __shfl_xor(e, o, 32);

    const int lab = (s < MAX_POS) ? 1 : 0;
    const int tgt = lab ? gt_labels[gtidx[roi]] : 0;  // background for negs
    const float ce = (m + __logf(e)) - row[tgt];

    if (lane == 0) {
      cls_acc += ce;
      nv += 1.0f;

      // ---- smooth-L1 regression (positives only) ----
      if (lab) {
        const float4 p = reinterpret_cast<const float4*>(rois)[roi];
        const float4 g = reinterpret_cast<const float4*>(gt)[gtidx[roi]];
        const float pw = p.z - p.x, ph = p.w - p.y;
        const float px = p.x + 0.5f * pw, py = p.y + 0.5f * ph;
        const float gw = g.z - g.x, gh = g.w - g.y;
        const float gx = g.x + 0.5f * gw, gy = g.y + 0.5f * gh;
        const float iw = 1.0f / fmaxf(pw, 1e-8f);
        const float ih = 1.0f / fmaxf(ph, 1e-8f);
        const float tx = (gx - px) * iw;
        const float ty = (gy - py) * ih;
        const float tw = __logf(fmaxf(gw, 1e-8f) / fmaxf(pw, 1e-8f));
        const float th = __logf(fmaxf(gh, 1e-8f) / fmaxf(ph, 1e-8f));
        const float* pr = pred + (size_t)roi * 4;
        const float d[4] = {pr[0] - tx, pr[1] - ty, pr[2] - tw, pr[3] - th};
        float sl = 0.0f;
#pragma unroll
        for (int c = 0; c < 4; ++c) {
          const float ad = fabsf(d[c]);
          sl += (ad < 1.0f) ? 0.5f * ad * ad : ad - 0.5f;
        }
        reg_acc += sl;
      }
    }
  }

  if (lane == 0) {
    partial[wave] = cls_acc;
    partial[8 + wave] = reg_acc;
    partial[16 + wave] = nv;
  }
  __syncthreads();
  if (threadIdx.x == 0) {
    float c = 0.0f, r = 0.0f, n = 0.0f;
    for (int w = 0; w < 8; ++w) {
      c += partial[w];
      r += partial[8 + w];
      n += partial[16 + w];
    }
    n = fmaxf(n, 1.0f);
    out[0] = c / n;           // roi_cls_loss
    out[1] = 2.0f * r / n;    // roi_reg_loss (REG_LOSS_WEIGHT=2)
  }
}

// ---------------------------------------------------------------------------
extern "C" void kernel_launch(void* const* d_in, const int* in_sizes, int n_in,
                              void* d_out, int out_size, void* d_ws,
                              size_t ws_size, hipStream_t stream) {
  const float* rois  = (const float*)d_in[0];  // [200000,4]
  const float* score = (const float*)d_in[1];  // [200000,81]
  const float* pred  = (const float*)d_in[2];  // [200000,4]
  const float* gt    = (const float*)d_in[3];  // [128,4]
  const int*   glab  = (const int*)d_in[4];    // [128]

  char* ws = (char*)d_ws;  // ~4.0 MB used
  u64*      poskey = (u64*)(ws);
  u64*      negkey = (u64*)(ws + 8ull * N_ROIS);
  unsigned* gtix   = (unsigned*)(ws + 16ull * N_ROIS);
  int*      sel    = (int*)(ws + 20ull * N_ROIS);
  int*      meta   = sel + N_SEL;

  iou_label_kernel<<<(N_ROIS + 255) / 256, 256, 0, stream>>>(
      rois, gt, poskey, negkey, gtix);
  select_kernel<<<1, 1024, 0, stream>>>(poskey, negkey, sel, meta);
  loss_kernel<<<1, 256, 0, stream>>>(rois, gt, score, pred, glab, gtix, sel,
                                     meta, (float*)d_out);
}